// MaskedHistLoss_old_22737556865704
// MI455X (gfx1250) — compile-verified
//
#include <hip/hip_runtime.h>

// Problem constants (reference: input [1,128,512,512] f32, mask [1,512,512],
// target_hist [128,256], target_min/max [128]).
#define C_CH   128
#define HW     (512*512)          // N per channel = 262144
#define NBINS  256
#define NBLK   64                 // streaming chunks per channel
#define BLK_EL (HW / NBLK)        // 4096 elements per chunk
#define KB     8192               // coarse sortable-key buckets (top 13 bits)
#define KB_SHIFT 19
#define FIXP   268435456.0f       // 2^28 per-element fixed-point scale
#define LOSSP  1048576.0          // 2^20 loss fixed-point scale

// d_out layout (tuple flattened): [loss(1), cur_hist(128*256), cur_min(128), cur_max(128)]
#define OUT_HIST 1
#define OUT_MIN  (1 + C_CH*NBINS)
#define OUT_MAX  (OUT_MIN + C_CH)

typedef float v2f __attribute__((ext_vector_type(2)));
typedef float v8f __attribute__((ext_vector_type(8)));
typedef unsigned int u32x4 __attribute__((ext_vector_type(4)));
typedef int i32x4 __attribute__((ext_vector_type(4)));
typedef int i32x8 __attribute__((ext_vector_type(8)));

// ---------------------------------------------------------------------------
// K0: zero the integer-atomic accumulators (ihist, loss accumulator).
// ---------------------------------------------------------------------------
__global__ void k0_init(unsigned int* __restrict__ ihist,
                        unsigned long long* __restrict__ lossAcc) {
  int i = blockIdx.x * blockDim.x + threadIdx.x;
  if (i < C_CH * NBINS) ihist[i] = 0u;
  if (i == 0) *lossAcc = 0ull;
}

// ---------------------------------------------------------------------------
// K1: per-(channel, chunk) partial min / max / sum(x^2) of x = mask*input.
// ---------------------------------------------------------------------------
__global__ __launch_bounds__(256)
void k1_partials(const float* __restrict__ input, const float* __restrict__ mask,
                 float* __restrict__ pmin, float* __restrict__ pmax,
                 float* __restrict__ psq) {
  __shared__ float smin[256], smax[256], ssq[256];
  const int c = blockIdx.y, blk = blockIdx.x, tid = threadIdx.x;
  const float4* xi4 = (const float4*)(input + (size_t)c * HW + (size_t)blk * BLK_EL);
  const float4* mk4 = (const float4*)(mask + (size_t)blk * BLK_EL);
  float mn = __FLT_MAX__, mx = -__FLT_MAX__, sq = 0.0f;
  for (int i = tid; i < BLK_EL / 4; i += 256) {
    float4 a = xi4[i], m = mk4[i];
    float x0 = a.x * m.x, x1 = a.y * m.y, x2 = a.z * m.z, x3 = a.w * m.w;
    mn = fminf(mn, fminf(fminf(x0, x1), fminf(x2, x3)));
    mx = fmaxf(mx, fmaxf(fmaxf(x0, x1), fmaxf(x2, x3)));
    sq += x0 * x0 + x1 * x1 + x2 * x2 + x3 * x3;
  }
  smin[tid] = mn; smax[tid] = mx; ssq[tid] = sq;
  __syncthreads();
  for (int s = 128; s > 0; s >>= 1) {
    if (tid < s) {
      smin[tid] = fminf(smin[tid], smin[tid + s]);
      smax[tid] = fmaxf(smax[tid], smax[tid + s]);
      ssq[tid] += ssq[tid + s];
    }
    __syncthreads();
  }
  if (tid == 0) {
    pmin[c * NBLK + blk] = smin[0];
    pmax[c * NBLK + blk] = smax[0];
    psq [c * NBLK + blk] = ssq[0];
  }
}

// ---------------------------------------------------------------------------
// K2: reduce NBLK partials per channel -> cur_min/cur_max (d_out), sumsq (ws).
// Fixed-order tree reduction => deterministic.
// ---------------------------------------------------------------------------
__global__ __launch_bounds__(64)
void k2_finalize(const float* __restrict__ pmin, const float* __restrict__ pmax,
                 const float* __restrict__ psq, float* __restrict__ out,
                 float* __restrict__ sumsq) {
  __shared__ float smin[64], smax[64], ssq[64];
  const int c = blockIdx.x, t = threadIdx.x;
  smin[t] = pmin[c * NBLK + t];
  smax[t] = pmax[c * NBLK + t];
  ssq [t] = psq [c * NBLK + t];
  __syncthreads();
  for (int s = 32; s > 0; s >>= 1) {
    if (t < s) {
      smin[t] = fminf(smin[t], smin[t + s]);
      smax[t] = fmaxf(smax[t], smax[t + s]);
      ssq[t] += ssq[t + s];
    }
    __syncthreads();
  }
  if (t == 0) {
    out[OUT_MIN + c] = smin[0];
    out[OUT_MAX + c] = smax[0];
    sumsq[c] = ssq[0];
  }
}

// ---------------------------------------------------------------------------
// K3: 256-bin per-channel histogram. The 16KB input chunk and 16KB mask chunk
// are staged into LDS with the CDNA5 async engine (GLOBAL_LOAD_ASYNC_TO_LDS_B128,
// tracked by ASYNCcnt), then binned out of LDS with integer ds atomics.
// ---------------------------------------------------------------------------
__global__ __launch_bounds__(256)
void k3_hist256(const float* __restrict__ input, const float* __restrict__ mask,
                const float* __restrict__ out, unsigned int* __restrict__ ihist) {
  __shared__ unsigned int h[NBINS];
  __shared__ float xin_s[BLK_EL];   // 16 KB
  __shared__ float msk_s[BLK_EL];   // 16 KB
  const int c = blockIdx.y, blk = blockIdx.x, tid = threadIdx.x;
  h[tid] = 0u;

  // Async stage: 4096 floats per array = 1024 B128 transfers = 256 lanes x 4.
  const char* gin = (const char*)(input + (size_t)c * HW + (size_t)blk * BLK_EL);
  const char* gmk = (const char*)(mask + (size_t)blk * BLK_EL);
  const unsigned ldsx = (unsigned)(uintptr_t)(void*)xin_s;
  const unsigned ldsm = (unsigned)(uintptr_t)(void*)msk_s;
#pragma unroll
  for (int it = 0; it < BLK_EL / (256 * 4); ++it) {      // 4 iterations
    const unsigned off = (unsigned)(it * 256 + tid) * 16u;
    asm volatile("global_load_async_to_lds_b128 %0, %1, off"
                 :: "v"(ldsx + off), "v"(gin + off) : "memory");
    asm volatile("global_load_async_to_lds_b128 %0, %1, off"
                 :: "v"(ldsm + off), "v"(gmk + off) : "memory");
  }
  asm volatile("s_wait_asynccnt 0" ::: "memory");
  __syncthreads();

  const float mn = out[OUT_MIN + c];
  const float mx = out[OUT_MAX + c];
  const float inv = 255.0f / fmaxf(mx - mn, 1e-8f);
  const float4* xs4 = (const float4*)xin_s;
  const float4* ms4 = (const float4*)msk_s;
  for (int i = tid; i < BLK_EL / 4; i += 256) {
    float4 a = xs4[i], m = ms4[i];
    float xs[4] = {a.x * m.x, a.y * m.y, a.z * m.z, a.w * m.w};
#pragma unroll
    for (int j = 0; j < 4; ++j) {
      int b = (int)((xs[j] - mn) * inv);       // trunc toward zero, matches astype(int32)
      b = min(NBINS - 1, max(0, b));
      atomicAdd(&h[b], 1u);
    }
  }
  __syncthreads();
  atomicAdd(&ihist[c * NBINS + tid], h[tid]);
}

// ---------------------------------------------------------------------------
// K4a: target-hist CDF via WMMA. cumsum over 256 bins == hist(16ch x 256) x L,
// L = lower-triangular ones. One wave per (16-channel group, 16-bin out tile);
// accumulate 4J all-ones K-chunks plus 4 triangular K-chunks with
// V_WMMA_F32_16X16X4_F32 (f32 WMMA is exact for counts < 2^24).
// ---------------------------------------------------------------------------
__global__ __launch_bounds__(32)
void k4a_cdf_wmma(const float* __restrict__ thist, float* __restrict__ cum) {
  const int grp  = blockIdx.x;   // 0..7 : group of 16 channels
  const int J    = blockIdx.y;   // 0..15: 16-bin output tile
  const int lane = threadIdx.x;  // 0..31
  const int half = lane >> 4;
  const int l15  = lane & 15;
  const int chA  = grp * 16 + l15;
  v8f acc = {};
  const int qmax = 4 * J + 4;
  for (int q = 0; q < qmax; ++q) {
    const int kx = 4 * q + 2 * half;
    v2f A, B;
    A.x = thist[chA * NBINS + kx];
    A.y = thist[chA * NBINS + kx + 1];
    if (q < 4 * J) {
      B.x = 1.0f; B.y = 1.0f;                     // fully-below chunk: all ones
    } else {
      const int n = 16 * J + l15;                 // triangular boundary chunk
      B.x = (kx     <= n) ? 1.0f : 0.0f;
      B.y = (kx + 1 <= n) ? 1.0f : 0.0f;
    }
    acc = __builtin_amdgcn_wmma_f32_16x16x4_f32(false, A, false, B,
                                                (short)0, acc, false, false);
  }
#pragma unroll
  for (int r = 0; r < 8; ++r) {
    const int ch  = grp * 16 + r + 8 * half;
    const int bin = 16 * J + l15;
    cum[ch * NBINS + bin] = acc[r];
  }
}

// ---------------------------------------------------------------------------
// K4b: scale cum -> cdf = cum/total*N ; also emit float cur_hist to d_out.
// ---------------------------------------------------------------------------
__global__ __launch_bounds__(256)
void k4b_scale(const float* __restrict__ cum, float* __restrict__ cdf,
               const unsigned int* __restrict__ ihist, float* __restrict__ out) {
  const int c = blockIdx.x, t = threadIdx.x;
  const float tot = cum[c * NBINS + (NBINS - 1)];
  cdf[c * NBINS + t] = cum[c * NBINS + t] * ((float)HW / tot);
  out[OUT_HIST + c * NBINS + t] = (float)ihist[c * NBINS + t];
}

// ---------------------------------------------------------------------------
// K5: per-channel loss without sorting. 8192-bucket sortable-key histogram in
// LDS (u32 count + i64 fixed-point value-sum => deterministic ds_add_u64),
// LDS prefix scan, then P(k)=sum of k smallest values at the 256 CDF
// breakpoints via binary search + within-bucket interpolation.
// The 1KB cdf row is staged into LDS by the Tensor Data Mover (D# descriptor,
// TENSORcnt), issued once by wave 0 (TDM ignores EXEC; uniform branch skips
// the other waves entirely).
// loss_c = sum(x^2) + sum_b [ m_b t_b^2 - 2 t_b (P(F_b)-P(F_{b-1})) ].
// ---------------------------------------------------------------------------
__global__ __launch_bounds__(1024)
void k5_loss(const float* __restrict__ input, const float* __restrict__ mask,
             const float* __restrict__ cdf, const float* __restrict__ tmin,
             const float* __restrict__ tmax, const float* __restrict__ sumsq,
             unsigned long long* __restrict__ lossAcc) {
  __shared__ unsigned int        cnt[KB];     // 32 KB
  __shared__ unsigned long long  fsum[KB];    // 64 KB
  __shared__ unsigned int        scnt[1024];  // 4 KB
  __shared__ unsigned long long  ssum[1024];  // 8 KB
  __shared__ double              dacc[NBINS]; // 2 KB
  __shared__ float               cdf_s[NBINS];// 1 KB
  __shared__ unsigned long long  totSfixed;
  const int c = blockIdx.x, tid = threadIdx.x;

  for (int i = tid; i < KB; i += 1024) { cnt[i] = 0u; fsum[i] = 0ull; }

  // --- TDM stage of this channel's cdf row (256 f32, 1 row, stride 256) ---
#if __has_builtin(__builtin_amdgcn_tensor_load_to_lds) && __has_builtin(__builtin_amdgcn_s_wait_tensorcnt)
  if (tid < 32) {
    const unsigned long long gaddr =
        (unsigned long long)(uintptr_t)(const void*)(cdf + (size_t)c * NBINS);
    u32x4 g0;
    g0[0] = 1u;                                    // count=1, user D#
    g0[1] = (unsigned)(uintptr_t)(void*)cdf_s;     // lds_addr
    g0[2] = (unsigned)gaddr;                       // global_addr[31:0]
    g0[3] = ((unsigned)(gaddr >> 32) & 0x01FFFFFFu) | (2u << 30); // addr[56:32], type=2
    i32x8 g1;
    g1[0] = 0x00020000;            // data_size=4B, no multicast, no pad/iter
    g1[1] = (int)(256u << 16);     // tensor_dim0 = 256 (bits 63:48)
    g1[2] = (int)(1u << 16);       // tensor_dim0 hi = 0, tensor_dim1 = 1
    g1[3] = (int)(256u << 16);     // tensor_dim1 hi = 0, tile_dim0 = 256
    g1[4] = 1;                     // tile_dim1 = 1, tile_dim2 = 0
    g1[5] = 256;                   // tensor_dim0_stride = 256
    g1[6] = 0;
    g1[7] = 0;                     // tensor_dim1_stride = 0 (single row)
    i32x4 z4 = {};
    i32x8 z8 = {};
    __builtin_amdgcn_tensor_load_to_lds(g0, g1, z4, z4, z8, 0);  // 6-arg (clang-23)
    __builtin_amdgcn_s_wait_tensorcnt(0);
  }
#else
  if (tid < NBINS) cdf_s[tid] = cdf[c * NBINS + tid];
#endif
  __syncthreads();

  // Pass: build coarse histogram of x = mask*input (keyed by sortable top-13 bits).
  const float4* xi4 = (const float4*)(input + (size_t)c * HW);
  const float4* mk4 = (const float4*)mask;
  for (int i = tid; i < HW / 4; i += 1024) {
    __builtin_prefetch(&xi4[i + 2048], 0, 1);  // global_prefetch_b8
    float4 a = xi4[i], m = mk4[i];
    float xs[4] = {a.x * m.x, a.y * m.y, a.z * m.z, a.w * m.w};
#pragma unroll
    for (int j = 0; j < 4; ++j) {
      const float x = xs[j];
      const unsigned int u = __float_as_uint(x);
      const unsigned int s = (u & 0x80000000u) ? ~u : (u | 0x80000000u);
      const unsigned int k = s >> KB_SHIFT;
      atomicAdd(&cnt[k], 1u);                                    // ds_add_u32
      const long long q = llrintf(x * FIXP);
      atomicAdd(&fsum[k], (unsigned long long)q);                // ds_add_u64
    }
  }
  __syncthreads();

  // Exclusive prefix scan over KB buckets (8 per thread + serial top scan).
  {
    const int base = tid * (KB / 1024);
    unsigned int cs = 0; unsigned long long fs = 0ull;
#pragma unroll
    for (int j = 0; j < KB / 1024; ++j) { cs += cnt[base + j]; fs += fsum[base + j]; }
    scnt[tid] = cs; ssum[tid] = fs;
  }
  __syncthreads();
  if (tid == 0) {
    unsigned int rc = 0; unsigned long long rs = 0ull;
    for (int i = 0; i < 1024; ++i) {
      unsigned int tc = scnt[i]; unsigned long long ts = ssum[i];
      scnt[i] = rc; ssum[i] = rs; rc += tc; rs += ts;
    }
    totSfixed = rs;
  }
  __syncthreads();
  {
    const int base = tid * (KB / 1024);
    unsigned int rc = scnt[tid]; unsigned long long rs = ssum[tid];
#pragma unroll
    for (int j = 0; j < KB / 1024; ++j) {
      unsigned int tc = cnt[base + j]; unsigned long long ts = fsum[base + j];
      cnt[base + j] = rc; fsum[base + j] = rs; rc += tc; rs += ts;
    }
  }
  __syncthreads();

  const double invFix = 1.0 / (double)FIXP;
  const unsigned long long totS = totSfixed;
  // P(k): sum of k smallest x values.
  auto Pk = [&](long long k) -> double {
    if (k <= 0) return 0.0;
    if (k >= (long long)HW) return (double)(long long)totS * invFix;
    int lo = 0, hi = KB - 1;
    while (lo < hi) {                           // largest t with cntEx[t] < k
      int mid = (lo + hi + 1) >> 1;
      if ((long long)cnt[mid] < k) lo = mid; else hi = mid - 1;
    }
    const unsigned int ce = cnt[lo];
    const unsigned int cn = (lo < KB - 1) ? cnt[lo + 1] : (unsigned int)HW;
    const unsigned long long se = fsum[lo];
    const unsigned long long sn = (lo < KB - 1) ? fsum[lo + 1] : totS;
    const double cb = (double)(cn - ce);
    const double sb = (double)(long long)(sn - se) * invFix;
    const double frac = (double)(k - (long long)ce) / cb;
    return (double)(long long)se * invFix + frac * sb;
  };

  if (tid < NBINS) {
    const int b = tid;
    const float tmn = tmin[c], tmx = tmax[c];
    const double tb = ((double)b / 255.0) * (double)(tmx - tmn) + (double)tmn;
    long long Fb, Fbm1;
    if (b == NBINS - 1) Fb = HW;
    else {
      double v = floor((double)cdf_s[b]);
      Fb = (long long)fmin(fmax(v, 0.0), (double)HW);
    }
    if (b == 0) Fbm1 = 0;
    else {
      double v = floor((double)cdf_s[b - 1]);
      Fbm1 = (long long)fmin(fmax(v, 0.0), (double)HW);
    }
    if (Fb < Fbm1) Fb = Fbm1;
    const double mb = (double)(Fb - Fbm1);
    const double Sb = Pk(Fb) - Pk(Fbm1);
    dacc[b] = mb * tb * tb - 2.0 * tb * Sb;
  }
  __syncthreads();
  for (int s = NBINS / 2; s > 0; s >>= 1) {
    if (tid < s) dacc[tid] += dacc[tid + s];
    __syncthreads();
  }
  if (tid == 0) {
    const double chLoss = (double)sumsq[c] + dacc[0];
    const long long fx = llrint(chLoss * LOSSP);
    atomicAdd(lossAcc, (unsigned long long)fx);   // integer atomic: deterministic
  }
}

// ---------------------------------------------------------------------------
// K6: finalize loss = 0.01 * accum / (C*N).
// ---------------------------------------------------------------------------
__global__ void k6_final(const unsigned long long* __restrict__ lossAcc,
                         float* __restrict__ out) {
  if (blockIdx.x == 0 && threadIdx.x == 0) {
    const double acc = (double)(long long)(*lossAcc) / LOSSP;
    out[0] = (float)(0.01 * acc / ((double)C_CH * (double)HW));
  }
}

// ---------------------------------------------------------------------------
extern "C" void kernel_launch(void* const* d_in, const int* in_sizes, int n_in,
                              void* d_out, int out_size, void* d_ws, size_t ws_size,
                              hipStream_t stream) {
  (void)in_sizes; (void)n_in; (void)out_size; (void)ws_size;
  const float* input = (const float*)d_in[0];   // [1,128,512,512]
  const float* mask  = (const float*)d_in[1];   // [1,512,512]
  const float* thist = (const float*)d_in[2];   // [128,256]
  const float* tmin  = (const float*)d_in[3];   // [128]
  const float* tmax  = (const float*)d_in[4];   // [128]
  float* out = (float*)d_out;

  // Workspace layout (≈0.5 MB total)
  char* w = (char*)d_ws;
  float* pmin  = (float*)(w + 0);                      // 128*64 f32
  float* pmax  = (float*)(w + 32768);
  float* psq   = (float*)(w + 65536);
  float* sumsq = (float*)(w + 98304);                  // 128 f32
  unsigned int* ihist = (unsigned int*)(w + 98816);    // 128*256 u32
  float* cum   = (float*)(w + 229888);                 // 128*256 f32
  float* cdf   = (float*)(w + 360960);                 // 128*256 f32
  unsigned long long* lossAcc = (unsigned long long*)(w + 492032);

  k0_init     <<<dim3(128),          dim3(256),  0, stream>>>(ihist, lossAcc);
  k1_partials <<<dim3(NBLK, C_CH),   dim3(256),  0, stream>>>(input, mask, pmin, pmax, psq);
  k2_finalize <<<dim3(C_CH),         dim3(64),   0, stream>>>(pmin, pmax, psq, out, sumsq);
  k3_hist256  <<<dim3(NBLK, C_CH),   dim3(256),  0, stream>>>(input, mask, out, ihist);
  k4a_cdf_wmma<<<dim3(8, 16),        dim3(32),   0, stream>>>(thist, cum);
  k4b_scale   <<<dim3(C_CH),         dim3(256),  0, stream>>>(cum, cdf, ihist, out);
  k5_loss     <<<dim3(C_CH),         dim3(1024), 0, stream>>>(input, mask, cdf, tmin, tmax,
                                                              sumsq, lossAcc);
  k6_final    <<<dim3(1),            dim3(1),    0, stream>>>(lossAcc, out);
}